// BilinearGridSample_2147483648104
// MI455X (gfx1250) — compile-verified
//
#include <hip/hip_runtime.h>

// Bilinear grid sample (mmcv semantics, align_corners=False).
// img:    (8, 64, 256, 256) f32
// points: (8, 384, 384, 2)  f32 in [-1, 1]
// out:    (8, 64, 384, 384) f32
//
// Memory-bound kernel: ~446 MB of HBM traffic -> ~19 us floor at 23.3 TB/s.
// Strategy: one thread per point; weights/offsets computed once and reused
// across all 64 channels; non-temporal stores keep the 134 MB image resident
// in the 192 MB L2 so the random gathers hit L2.

namespace {

constexpr int N_  = 8;
constexpr int C_  = 64;
constexpr int H_  = 256;
constexpr int W_  = 256;
constexpr int GH_ = 384;
constexpr int GW_ = 384;
constexpr int P_  = GH_ * GW_;   // 147456 points per image
constexpr int HW_ = H_ * W_;     // 65536 texels per plane

// Clang-native vector type (HIP's float2 is a class and is rejected by
// __builtin_nontemporal_load).
typedef float v2f __attribute__((ext_vector_type(2)));

__global__ __launch_bounds__(256)
void bilinear_grid_sample_kernel(const float* __restrict__ img,
                                 const float* __restrict__ pts,
                                 float* __restrict__ out)
{
    const int p = blockIdx.x * blockDim.x + threadIdx.x;  // point index in image
    const int n = blockIdx.y;                              // batch index
    if (p >= P_) return;

    // Coalesced 8-byte load of (x, y); points are streamed once -> non-temporal.
    const v2f* pv = reinterpret_cast<const v2f*>(pts) + (size_t)n * P_ + p;
    const v2f g = __builtin_nontemporal_load(pv);

    // Unnormalize (align_corners = False).
    const float x = ((g.x + 1.0f) * (float)W_ - 1.0f) * 0.5f;
    const float y = ((g.y + 1.0f) * (float)H_ - 1.0f) * 0.5f;

    const float x0f = floorf(x);
    const float y0f = floorf(y);
    const float fx = x - x0f;    // == x - x0, and (x1 - x) == 1 - fx
    const float fy = y - y0f;

    const int x0 = (int)x0f;
    const int y0 = (int)y0f;
    const int x1 = x0 + 1;
    const int y1 = y0 + 1;

    // mmcv pads the image with a zero border and clamps indices into the pad:
    // an out-of-bounds corner contributes exactly zero. Equivalent transform:
    // zero that corner's weight and clamp its index in-bounds (branch-free loop).
    float wxl = 1.0f - fx;
    float wxh = fx;
    float wyl = 1.0f - fy;
    float wyh = fy;
    if (x0 < 0 || x0 >= W_) wxl = 0.0f;
    if (x1 < 0 || x1 >= W_) wxh = 0.0f;
    if (y0 < 0 || y0 >= H_) wyl = 0.0f;
    if (y1 < 0 || y1 >= H_) wyh = 0.0f;

    const float wa = wxl * wyl;  // (y0, x0)
    const float wb = wxl * wyh;  // (y1, x0)
    const float wc = wxh * wyl;  // (y0, x1)
    const float wd = wxh * wyh;  // (y1, x1)

    const int cx0 = min(max(x0, 0), W_ - 1);
    const int cx1 = min(max(x1, 0), W_ - 1);
    const int cy0 = min(max(y0, 0), H_ - 1);
    const int cy1 = min(max(y1, 0), H_ - 1);

    const int offA = cy0 * W_ + cx0;
    const int offB = cy1 * W_ + cx0;
    const int offC = cy0 * W_ + cx1;
    const int offD = cy1 * W_ + cx1;

    const float* __restrict__ base  = img + (size_t)n * C_ * HW_;
    float* __restrict__       obase = out + (size_t)n * C_ * P_ + p;

    // Channel loop: 4 L2-resident gathers + 4 FMAs + 1 streaming store.
    // Unroll-by-8 keeps 32 loads in flight per wave for latency hiding.
    #pragma unroll 8
    for (int c = 0; c < C_; ++c) {
        const float* __restrict__ pc = base + (size_t)c * HW_;
        const float Ia = pc[offA];
        const float Ib = pc[offB];
        const float Ic = pc[offC];
        const float Id = pc[offD];
        const float v = Ia * wa + Ib * wb + Ic * wc + Id * wd;
        // Non-temporal: 302 MB of output must not evict the image from L2.
        __builtin_nontemporal_store(v, obase + (size_t)c * P_);
    }
}

} // namespace

extern "C" void kernel_launch(void* const* d_in, const int* in_sizes, int n_in,
                              void* d_out, int out_size, void* d_ws, size_t ws_size,
                              hipStream_t stream) {
    const float* img = (const float*)d_in[0];   // (8, 64, 256, 256)
    const float* pts = (const float*)d_in[1];   // (8, 384, 384, 2)
    float* out = (float*)d_out;                 // (8, 64, 384, 384)

    dim3 block(256, 1, 1);
    dim3 grid(P_ / 256, N_, 1);                 // 576 x 8 blocks, 8 wave32/WG
    bilinear_grid_sample_kernel<<<grid, block, 0, stream>>>(img, pts, out);
}